// DiscreteMeasure_GaussianNet_12472585028219
// MI455X (gfx1250) — compile-verified
//
#include <hip/hip_runtime.h>
#include <hip/hip_bf16.h>
#include <math.h>

typedef __attribute__((ext_vector_type(2))) float v2f;
typedef __attribute__((ext_vector_type(8))) float v8f;

#define B_ 16
#define N_ 10000
#define D_ 64
#define R_ 64
#define H_ 512
#define K_ 32

// ---------------- s2[n] = ||support[n]||^2 ----------------
__global__ void s2_kernel(const float* __restrict__ S, float* __restrict__ s2) {
    int n = blockIdx.x * blockDim.x + threadIdx.x;
    if (n >= N_) return;
    const float* row = S + (size_t)n * D_;
    float acc = 0.f;
#pragma unroll
    for (int d = 0; d < D_; ++d) acc += row[d] * row[d];
    s2[n] = acc;
}

// ---------------- top-k per batch (32 argmax passes over LDS copy) ----------------
__global__ void topk_kernel(const float* __restrict__ E,
                            float* __restrict__ tw, int* __restrict__ ti) {
    __shared__ float vals[N_];
    __shared__ float rv[256];
    __shared__ int   ri[256];
    int b = blockIdx.x, tid = threadIdx.x;
    for (int i = tid; i < N_; i += 256) vals[i] = E[(size_t)b * N_ + i];
    __syncthreads();
    for (int sel = 0; sel < K_; ++sel) {
        float bv = -INFINITY; int bi = 0x7fffffff;
        for (int i = tid; i < N_; i += 256) {
            float v = vals[i];
            if (v > bv || (v == bv && i < bi)) { bv = v; bi = i; }
        }
        rv[tid] = bv; ri[tid] = bi;
        __syncthreads();
        for (int off = 128; off > 0; off >>= 1) {
            if (tid < off) {
                float ov = rv[tid + off]; int oi = ri[tid + off];
                if (ov > rv[tid] || (ov == rv[tid] && oi < ri[tid])) { rv[tid] = ov; ri[tid] = oi; }
            }
            __syncthreads();
        }
        if (tid == 0) {
            tw[b * K_ + sel] = rv[0];
            ti[b * K_ + sel] = ri[0];
            vals[ri[0]] = -INFINITY;
        }
        __syncthreads();
    }
}

// ---------------- MLP only on the B*K selected rows ----------------
// block = 64 threads, grid = B_*K_ ; produces top_mean[row][64] and m2[row]
__global__ void mlp_kernel(const float* __restrict__ Remb, const float* __restrict__ S,
                           const float* __restrict__ W1, const float* __restrict__ B1,
                           const float* __restrict__ W2, const float* __restrict__ B2,
                           const int* __restrict__ ti,
                           float* __restrict__ TM, float* __restrict__ M2) {
    __shared__ float h[H_];
    __shared__ float red[D_];
    int row = blockIdx.x;          // b*K_ + j
    int b = row / K_;
    int n = ti[row];
    int t = threadIdx.x;           // 0..63
    const float* rrow = Remb + (size_t)b * R_;
    const float* srow = S + (size_t)n * D_;
#pragma unroll
    for (int c = 0; c < H_ / 64; ++c) {
        int j = t + 64 * c;
        const float* wrow = W1 + (size_t)j * (R_ + D_);
        float a = B1[j];
        for (int r = 0; r < R_; ++r) a += rrow[r] * wrow[r];
        for (int d = 0; d < D_; ++d) a += srow[d] * wrow[R_ + d];
        h[j] = fmaxf(a, 0.f);
    }
    __syncthreads();
    int d = t;
    float m = B2[d];
    const float* w2row = W2 + (size_t)d * H_;
    for (int j = 0; j < H_; ++j) m += h[j] * w2row[j];
    TM[(size_t)row * D_ + d] = m;
    red[d] = m * m;
    __syncthreads();
    for (int off = 32; off > 0; off >>= 1) {
        if (t < off) red[t] += red[t + off];
        __syncthreads();
    }
    if (t == 0) M2[row] = red[0];
}

// ---------------- fused WMMA cross + exp + weighted k-sum ----------------
// one wave32 per (batch, 16-row N tile); 2 accumulators cover k=0..15 and k=16..31
__global__ void cross_kernel(const float* __restrict__ S, const float* __restrict__ TM,
                             const float* __restrict__ s2, const float* __restrict__ M2,
                             const float* __restrict__ TW, float* __restrict__ tmp) {
    int wave = (blockIdx.x * blockDim.x + threadIdx.x) >> 5;
    int lane = threadIdx.x & 31;
    int b  = wave / (N_ / 16);
    int n0 = (wave % (N_ / 16)) * 16;

    int m    = lane & 15;            // A: row M within tile / B: column (k component)
    int koff = (lane >> 4) * 2;      // lanes 16-31 carry K=2,3 of each 4-wide K step

    const float* srow  = S  + (size_t)(n0 + m) * D_;
    const float* brow0 = TM + (size_t)(b * K_ + m) * D_;
    const float* brow1 = TM + (size_t)(b * K_ + 16 + m) * D_;

    v8f acc0 = {}; v8f acc1 = {};
#pragma unroll
    for (int kk = 0; kk < D_ / 4; ++kk) {
        int K = kk * 4 + koff;
        v2f a  = *(const v2f*)(srow + K);
        v2f b0 = *(const v2f*)(brow0 + K);
        v2f b1 = *(const v2f*)(brow1 + K);
        acc0 = __builtin_amdgcn_wmma_f32_16x16x4_f32(false, a, false, b0, (short)0, acc0, false, false);
        acc1 = __builtin_amdgcn_wmma_f32_16x16x4_f32(false, a, false, b1, (short)0, acc1, false, false);
    }

    float m2_0 = M2[b * K_ + m],      m2_1 = M2[b * K_ + 16 + m];
    float w0   = TW[b * K_ + m],      w1   = TW[b * K_ + 16 + m];
    int rbase = n0 + ((lane >> 4) << 3);   // C layout: VGPR i -> row i (+8 for hi lanes)

    float sum[8];
#pragma unroll
    for (int i = 0; i < 8; ++i) {
        float s2r = s2[rbase + i];
        float t0 = acc0[i] - 0.5f * (s2r + m2_0);   // always <= 0 -> exp safe
        float t1 = acc1[i] - 0.5f * (s2r + m2_1);
        sum[i] = w0 * __expf(t0) + w1 * __expf(t1);
    }
#pragma unroll
    for (int i = 0; i < 8; ++i) {
#pragma unroll
        for (int off = 8; off > 0; off >>= 1)
            sum[i] += __shfl_xor(sum[i], off, 16);  // reduce 16 k-columns
    }
    if ((lane & 15) == 0) {
#pragma unroll
        for (int i = 0; i < 8; ++i)
            tmp[(size_t)b * N_ + rbase + i] = sum[i];
    }
}

// ---------------- per-batch L1 normalization ----------------
__global__ void norm_kernel(const float* __restrict__ tmp, float* __restrict__ out) {
    __shared__ float red[256];
    int b = blockIdx.x, t = threadIdx.x;
    float s = 0.f;
    for (int i = t; i < N_; i += 256) s += tmp[(size_t)b * N_ + i];
    red[t] = s;
    __syncthreads();
    for (int off = 128; off > 0; off >>= 1) {
        if (t < off) red[t] += red[t + off];
        __syncthreads();
    }
    float inv = 1.0f / (red[0] + 1e-10f);
    for (int i = t; i < N_; i += 256) out[(size_t)b * N_ + i] = tmp[(size_t)b * N_ + i] * inv;
}

extern "C" void kernel_launch(void* const* d_in, const int* in_sizes, int n_in,
                              void* d_out, int out_size, void* d_ws, size_t ws_size,
                              hipStream_t stream) {
    const float* E  = (const float*)d_in[0];
    const float* Re = (const float*)d_in[1];
    const float* S  = (const float*)d_in[2];
    const float* W1 = (const float*)d_in[3];
    const float* B1 = (const float*)d_in[4];
    const float* W2 = (const float*)d_in[5];
    const float* B2 = (const float*)d_in[6];

    char* ws = (char*)d_ws;
    int*   ti  = (int*)  (ws + 0);        // 512 ints
    float* tw  = (float*)(ws + 2048);     // 512 f
    float* M2  = (float*)(ws + 4096);     // 512 f
    float* s2  = (float*)(ws + 6144);     // 10000 f
    float* TM  = (float*)(ws + 46336);    // 512*64 f
    float* tmp = (float*)(ws + 177408);   // 16*10000 f
    float* out = (float*)d_out;

    s2_kernel  <<<(N_ + 255) / 256, 256, 0, stream>>>(S, s2);
    topk_kernel<<<B_, 256, 0, stream>>>(E, tw, ti);
    mlp_kernel <<<B_ * K_, 64, 0, stream>>>(Re, S, W1, B1, W2, B2, ti, TM, M2);
    int waves = B_ * (N_ / 16);                    // 10000 tiles, one wave each
    cross_kernel<<<waves / 8, 256, 0, stream>>>(S, TM, s2, M2, tw, tmp);
    norm_kernel <<<B_, 256, 0, stream>>>(tmp, out);
}